// Attention_69509750718795
// MI455X (gfx1250) — compile-verified
//
#include <hip/hip_runtime.h>
#include <math.h>

#define B_ 2
#define N_ 1024
#define C_ 1024
#define H_ 16
#define HD_ 64
#define PAST_ 8192
#define NTOT_ (PAST_ + N_)          // 9216
#define ANCHOR_ 512
#define NC_ (NTOT_ - ANCHOR_)       // 8704
#define BUDGET_ 4000
#define KEEP_ (BUDGET_ - ANCHOR_)   // 3488
#define NOLD_ (NC_ - N_)            // 7680
#define SCALE_ 0.125f
#define IMPW_ 0.5f

typedef __attribute__((ext_vector_type(2))) float v2f;
typedef __attribute__((ext_vector_type(8))) float v8f;

// ---------------------------------------------------------------------------
// GEMM: C[m][n] = sum_k A[m][k] * W[n][k] + bias[n]
// mode 0: out row-major (M x Nout).  mode 1: QKV scatter into (B,H,N,HD).
// One wave32 per 16x16 tile, v_wmma_f32_16x16x4_f32, K stepped by 4.
// ---------------------------------------------------------------------------
__global__ void k_gemm_nt(const float* __restrict__ A, const float* __restrict__ W,
                          const float* __restrict__ bias,
                          float* __restrict__ out,
                          float* __restrict__ qd, float* __restrict__ kd, float* __restrict__ vd,
                          int M, int Nout, int K, int mode)
{
    int lane = threadIdx.x & 31;
    int wid  = blockIdx.x * (blockDim.x >> 5) + (threadIdx.x >> 5);
    int tilesN = Nout >> 4;
    int tm = wid / tilesN;
    int tn = wid % tilesN;
    if (tm >= (M >> 4)) return;

    int half = lane >> 4;          // 0: K+0/1, 1: K+2/3
    int l16  = lane & 15;
    int koff = half << 1;
    int mrow = tm * 16 + l16;      // A row this lane feeds
    int ncol = tn * 16 + l16;      // W row (= output column) this lane feeds

    const float* arow = A + (size_t)mrow * K + koff;
    const float* wrow = W + (size_t)ncol * K + koff;

    v8f acc = {};
    for (int k0 = 0; k0 < K; k0 += 4) {
        if ((k0 & 63) == 0) {                         // stream-ahead: global_prefetch_b8
            __builtin_prefetch(arow + k0 + 64, 0, 1);
            __builtin_prefetch(wrow + k0 + 64, 0, 1);
        }
        v2f a; a.x = arow[k0]; a.y = arow[k0 + 1];
        v2f b; b.x = wrow[k0]; b.y = wrow[k0 + 1];
        acc = __builtin_amdgcn_wmma_f32_16x16x4_f32(false, a, false, b, (short)0, acc, false, false);
    }

    int colg = tn * 16 + l16;
    float bv = bias[colg];
    if (mode == 0) {
        for (int r = 0; r < 8; ++r) {
            int mg = tm * 16 + r + 8 * half;
            out[(size_t)mg * Nout + colg] = acc[r] + bv;
        }
    } else {
        int sec = colg >> 10;            // 0=q 1=k 2=v
        int h   = (colg & 1023) >> 6;
        int d   = colg & 63;
        float* dst = (sec == 0) ? qd : ((sec == 1) ? kd : vd);
        for (int r = 0; r < 8; ++r) {
            int mg = tm * 16 + r + 8 * half;    // global row in [0, B*N)
            int bb = mg >> 10;
            int n  = mg & 1023;
            dst[(((size_t)bb * H_ + h) * N_ + n) * HD_ + d] = acc[r] + bv;
        }
    }
}

// ---------------------------------------------------------------------------
// Mean of L2-normalized candidate keys per (b,h).  One block per (b,h),
// 8 waves; each wave owns a token per iteration, 32 lanes cover 64 dims (2 ea).
// ---------------------------------------------------------------------------
__global__ void k_knmean(const float* __restrict__ past_k, const float* __restrict__ knew,
                         float* __restrict__ knmean)
{
    int bh = blockIdx.x;                 // 0..31
    int b = bh / H_, h = bh % H_;
    int lane = threadIdx.x & 31;
    int wv   = threadIdx.x >> 5;         // 0..7
    __shared__ float red[8 * 64];

    const float* pk = past_k + ((size_t)b * H_ + h) * PAST_ * HD_;
    const float* nk = knew   + ((size_t)b * H_ + h) * N_ * HD_;
    int d = lane * 2;
    float ax = 0.f, ay = 0.f;
    for (int c = wv; c < NC_; c += 8) {
        int t = ANCHOR_ + c;
        const float* row = (t < PAST_) ? (pk + (size_t)t * HD_)
                                       : (nk + (size_t)(t - PAST_) * HD_);
        float x0 = row[d], x1 = row[d + 1];
        float ss = x0 * x0 + x1 * x1;
        for (int m = 1; m < 32; m <<= 1) ss += __shfl_xor(ss, m, 32);
        float inv = 1.0f / fmaxf(sqrtf(ss), 1e-12f);
        ax += x0 * inv; ay += x1 * inv;
    }
    red[wv * 64 + d]     = ax;
    red[wv * 64 + d + 1] = ay;
    __syncthreads();
    if (threadIdx.x < 64) {
        float s = 0.f;
        for (int w = 0; w < 8; ++w) s += red[w * 64 + threadIdx.x];
        knmean[(size_t)bh * HD_ + threadIdx.x] = s * (1.0f / (float)NC_);
    }
}

// ---------------------------------------------------------------------------
// div_avg[b][c] = mean_h (1 - dot(normalize(k[b,h,c]), mean[b,h]))   (c < NOLD_,
// all old candidates live in past_k).  One wave per candidate.
// ---------------------------------------------------------------------------
__global__ void k_divavg(const float* __restrict__ past_k, const float* __restrict__ knmean,
                         float* __restrict__ divavg)
{
    int lane = threadIdx.x & 31;
    int wid  = blockIdx.x * (blockDim.x >> 5) + (threadIdx.x >> 5);
    int b = wid / NOLD_;
    int c = wid % NOLD_;
    if (b >= B_) return;
    int t = ANCHOR_ + c;
    int d = lane * 2;
    float accum = 0.f;
    for (int h = 0; h < H_; ++h) {
        const float* row = past_k + (((size_t)b * H_ + h) * PAST_ + t) * HD_;
        const float* mv  = knmean + ((size_t)b * H_ + h) * HD_;
        float x0 = row[d], x1 = row[d + 1];
        float ss = x0 * x0 + x1 * x1;
        float dp = x0 * mv[d] + x1 * mv[d + 1];
        for (int m = 1; m < 32; m <<= 1) {
            ss += __shfl_xor(ss, m, 32);
            dp += __shfl_xor(dp, m, 32);
        }
        float inv = 1.0f / fmaxf(sqrtf(ss), 1e-12f);
        accum += 1.0f - dp * inv;
    }
    if (lane == 0) divavg[(size_t)b * NOLD_ + c] = accum * (1.0f / (float)H_);
}

// ---------------------------------------------------------------------------
// Per-batch min/max of div_avg (old) and importance (new).
// ---------------------------------------------------------------------------
__global__ void k_minmax(const float* __restrict__ divavg, const float* __restrict__ imp,
                         float* __restrict__ minmax)
{
    int b = blockIdx.x;
    __shared__ float smn[256], smx[256];
    // --- old ---
    float mn = 3.4e38f, mx = -3.4e38f;
    for (int i = threadIdx.x; i < NOLD_; i += 256) {
        float v = divavg[(size_t)b * NOLD_ + i];
        mn = fminf(mn, v); mx = fmaxf(mx, v);
    }
    smn[threadIdx.x] = mn; smx[threadIdx.x] = mx;
    __syncthreads();
    for (int s = 128; s > 0; s >>= 1) {
        if (threadIdx.x < s) {
            smn[threadIdx.x] = fminf(smn[threadIdx.x], smn[threadIdx.x + s]);
            smx[threadIdx.x] = fmaxf(smx[threadIdx.x], smx[threadIdx.x + s]);
        }
        __syncthreads();
    }
    if (threadIdx.x == 0) { minmax[b * 8 + 0] = smn[0]; minmax[b * 8 + 1] = smx[0]; }
    __syncthreads();
    // --- new (importance) ---
    mn = 3.4e38f; mx = -3.4e38f;
    for (int i = threadIdx.x; i < N_; i += 256) {
        float v = imp[(size_t)b * N_ + i];
        mn = fminf(mn, v); mx = fmaxf(mx, v);
    }
    smn[threadIdx.x] = mn; smx[threadIdx.x] = mx;
    __syncthreads();
    for (int s = 128; s > 0; s >>= 1) {
        if (threadIdx.x < s) {
            smn[threadIdx.x] = fminf(smn[threadIdx.x], smn[threadIdx.x + s]);
            smx[threadIdx.x] = fmaxf(smx[threadIdx.x], smx[threadIdx.x + s]);
        }
        __syncthreads();
    }
    if (threadIdx.x == 0) { minmax[b * 8 + 2] = smn[0]; minmax[b * 8 + 3] = smx[0]; }
}

// ---------------------------------------------------------------------------
// combined[b][c] : min-max normalized diversity / importance blend.
// ---------------------------------------------------------------------------
__global__ void k_comb(const float* __restrict__ divavg, const float* __restrict__ imp,
                       const float* __restrict__ minmax, float* __restrict__ comb)
{
    int idx = blockIdx.x * 256 + threadIdx.x;
    int b = idx / NC_, c = idx % NC_;
    if (b >= B_) return;
    float v;
    if (c < NOLD_) {
        float mn = minmax[b * 8 + 0], mx = minmax[b * 8 + 1];
        v = (1.0f - IMPW_) * (divavg[(size_t)b * NOLD_ + c] - mn) / (mx - mn + 1e-8f);
    } else {
        float mn = minmax[b * 8 + 2], mx = minmax[b * 8 + 3];
        v = IMPW_ * (imp[(size_t)b * N_ + (c - NOLD_)] - mn) / (mx - mn + 1e-8f);
    }
    comb[(size_t)b * NC_ + c] = v;
}

// ---------------------------------------------------------------------------
// Exact top-k by rank counting (stable tie-break on index == jax top_k).
// flag[b][i] = 1  iff element i is among the KEEP_ largest of combined[b].
// ---------------------------------------------------------------------------
__global__ void k_rank(const float* __restrict__ comb, int* __restrict__ flag)
{
    int b = blockIdx.x / (NC_ / 256);
    int i = (blockIdx.x % (NC_ / 256)) * 256 + threadIdx.x;
    const float* cb = comb + (size_t)b * NC_;
    float vi = cb[i];
    __shared__ float tile[256];
    int rank = 0;
    for (int j0 = 0; j0 < NC_; j0 += 256) {
        __syncthreads();
        tile[threadIdx.x] = cb[j0 + threadIdx.x];
        __syncthreads();
        for (int jj = 0; jj < 256; ++jj) {
            float vj = tile[jj];
            int j = j0 + jj;
            rank += (vj > vi) || (vj == vi && j < i);
        }
    }
    flag[(size_t)b * NC_ + i] = (rank < KEEP_) ? 1 : 0;
}

// ---------------------------------------------------------------------------
// Stream compaction of kept indices in ascending order (== sorted top-k idx).
// One wave32 per batch: ballot + prefix popcount.
// ---------------------------------------------------------------------------
__global__ void k_compact(const int* __restrict__ flag, int* __restrict__ gidx)
{
    int b = blockIdx.x;
    int lane = threadIdx.x;
    const int* f = flag + (size_t)b * NC_;
    int* g = gidx + (size_t)b * KEEP_;
    int base = 0;
    for (int c0 = 0; c0 < NC_; c0 += 32) {
        int fl = f[c0 + lane];
        unsigned long long bm = __ballot(fl != 0);
        unsigned mask = (unsigned)bm;
        int pre = __popc(mask & ((1u << lane) - 1u));
        int pos = base + pre;
        if (fl && pos < KEEP_) g[pos] = c0 + lane;
        base += __popc(mask);
    }
}

// ---------------------------------------------------------------------------
// Gather kept K/V rows into compact (B,H,KEEP_,HD) caches.  float4 per thread.
// ---------------------------------------------------------------------------
__global__ void k_gather(const float* __restrict__ past_k, const float* __restrict__ past_v,
                         const float* __restrict__ knew, const float* __restrict__ vnew,
                         const int* __restrict__ gidx,
                         float* __restrict__ kkeep, float* __restrict__ vkeep)
{
    size_t idx = (size_t)blockIdx.x * 256 + threadIdx.x;
    int d4 = (int)(idx & 15);
    size_t t1 = idx >> 4;
    int j  = (int)(t1 % KEEP_);
    size_t t2 = t1 / KEEP_;
    int h = (int)(t2 % H_);
    int b = (int)(t2 / H_);
    if (b >= B_) return;
    int c = gidx[(size_t)b * KEEP_ + j];
    int t = ANCHOR_ + c;
    const float4 *ks, *vs;
    if (t < PAST_) {
        ks = (const float4*)(past_k + (((size_t)b * H_ + h) * PAST_ + t) * HD_);
        vs = (const float4*)(past_v + (((size_t)b * H_ + h) * PAST_ + t) * HD_);
    } else {
        ks = (const float4*)(knew + (((size_t)b * H_ + h) * N_ + (t - PAST_)) * HD_);
        vs = (const float4*)(vnew + (((size_t)b * H_ + h) * N_ + (t - PAST_)) * HD_);
    }
    float4* kdst = (float4*)(kkeep + (((size_t)b * H_ + h) * KEEP_ + j) * HD_);
    float4* vdst = (float4*)(vkeep + (((size_t)b * H_ + h) * KEEP_ + j) * HD_);
    kdst[d4] = ks[d4];
    vdst[d4] = vs[d4];
}

// ---------------------------------------------------------------------------
// Flash attention over the 4000-entry kept cache.  One wave per (b,h,16-query
// tile).  S = Q·K^T and O += P·V via v_wmma_f32_16x16x4_f32; online softmax
// with half-wave shfl_xor row reductions; P restaged through per-wave LDS.
// ---------------------------------------------------------------------------
__global__ void k_attn(const float* __restrict__ q, const float* __restrict__ past_k,
                       const float* __restrict__ past_v,
                       const float* __restrict__ kkeep, const float* __restrict__ vkeep,
                       float* __restrict__ out)
{
    __shared__ float pls[4][256];          // 16x16 P tile per wave
    int lane = threadIdx.x & 31;
    int wv   = threadIdx.x >> 5;
    int wid  = blockIdx.x * 4 + wv;        // 0..2047
    int qt = wid & 63;                     // query tile (N/16)
    int bh = wid >> 6;
    int h = bh & 15;
    int b = bh >> 4;
    int half = lane >> 4, l16 = lane & 15;
    int koff = half << 1;
    float* myP = &pls[wv][0];

    // Q fragments: full 16x64 tile striped as 16 x v2f per lane.
    const float* qrow = q + (((size_t)b * H_ + h) * N_ + (size_t)qt * 16 + l16) * HD_;
    v2f qa[16];
    for (int s = 0; s < 16; ++s) { qa[s].x = qrow[4 * s + koff]; qa[s].y = qrow[4 * s + koff + 1]; }

    float mrow[8], lrow[8];
    v8f o0 = {}, o1 = {}, o2 = {}, o3 = {};
    for (int r = 0; r < 8; ++r) { mrow[r] = -3.4e38f; lrow[r] = 0.f; }

    const float* kanch = past_k + ((size_t)b * H_ + h) * PAST_ * HD_;
    const float* vanch = past_v + ((size_t)b * H_ + h) * PAST_ * HD_;
    const float* kk = kkeep + ((size_t)b * H_ + h) * KEEP_ * HD_;
    const float* vk = vkeep + ((size_t)b * H_ + h) * KEEP_ * HD_;

    for (int tile = 0; tile < BUDGET_ / 16; ++tile) {
        int kbase = tile * 16;
        const float *Kt, *Vt;
        if (kbase < ANCHOR_) { Kt = kanch + (size_t)kbase * HD_; Vt = vanch + (size_t)kbase * HD_; }
        else                 { Kt = kk + (size_t)(kbase - ANCHOR_) * HD_; Vt = vk + (size_t)(kbase - ANCHOR_) * HD_; }

        // ---- S = Q @ Kt^T (16 wmma, K stepped by 4) ----
        v8f s = {};
        const float* krow = Kt + (size_t)l16 * HD_ + koff;   // key row n = l16
        for (int st = 0; st < 16; ++st) {
            v2f bf; bf.x = krow[4 * st]; bf.y = krow[4 * st + 1];
            s = __builtin_amdgcn_wmma_f32_16x16x4_f32(false, qa[st], false, bf, (short)0, s, false, false);
        }

        // ---- online softmax (row stats per half-wave) ----
        float tmax[8];
        for (int r = 0; r < 8; ++r) { s[r] *= SCALE_; tmax[r] = s[r]; }
        for (int m = 1; m < 16; m <<= 1)
            for (int r = 0; r < 8; ++r) tmax[r] = fmaxf(tmax[r], __shfl_xor(tmax[r], m, 32));
        float corr[8];
        for (int r = 0; r < 8; ++r) {
            float mnew = fmaxf(mrow[r], tmax[r]);
            corr[r] = __expf(mrow[r] - mnew);
            mrow[r] = mnew;
        }
        float rsum[8];
        for (int r = 0; r < 8; ++r) { float p = __expf(s[r] - mrow[r]); s[r] = p; rsum[r] = p; }
        for (int m = 1; m < 16; m <<= 1)
            for (int r = 0; r < 8; ++r) rsum[r] += __shfl_xor(rsum[r], m, 32);
        for (int r = 0; r < 8; ++r) lrow[r] = lrow[r] * corr[r] + rsum[r];
        for (int r = 0; r < 8; ++r) { o0[r] *= corr[r]; o1[r] *= corr[r]; o2[r] *= corr[r]; o3[r] *= corr[r]; }

        // ---- restage P: D-layout -> LDS [m][n] -> A-layout fragments ----
        __syncthreads();
        for (int r = 0; r < 8; ++r) myP[(r + 8 * half) * 16 + l16] = s[r];
        __syncthreads();

        // ---- O += P @ Vt (4 n-tiles x 4 K-steps = 16 wmma) ----
        for (int st = 0; st < 4; ++st) {
            v2f af; af.x = myP[l16 * 16 + 4 * st + koff]; af.y = myP[l16 * 16 + 4 * st + koff + 1];
            const float* v0 = Vt + (size_t)(4 * st + koff) * HD_;
            const float* v1 = v0 + HD_;
            v2f b0; b0.x = v0[l16];      b0.y = v1[l16];
            v2f b1; b1.x = v0[16 + l16]; b1.y = v1[16 + l16];
            v2f b2; b2.x = v0[32 + l16]; b2.y = v1[32 + l16];
            v2f b3; b3.x = v0[48 + l16]; b3.y = v1[48 + l16];
            o0 = __builtin_amdgcn_wmma_f32_16x16x4_f32(false, af, false, b0, (short)0, o0, false, false);
            o1 = __builtin_amdgcn_wmma_f32_16x16x4_f32(false, af, false, b1, (short)0, o1, false, false);
            o2 = __builtin_amdgcn_wmma_f32_16x16x4_f32(false, af, false, b2, (short)0, o2, false, false);
            o3 = __builtin_amdgcn_wmma_f32_16x16x4_f32(false, af, false, b3, (short)0, o3, false, false);
        }
    }

    // ---- normalize and write (B,N,C) with head transpose fused ----
    for (int r = 0; r < 8; ++r) {
        int m = qt * 16 + r + 8 * half;
        float inv = 1.0f / lrow[r];
        size_t base = ((size_t)b * N_ + m) * C_ + (size_t)h * HD_;
        out[base +  0 + l16] = o0[r] * inv;
        out[base + 16 + l16] = o1[r] * inv;
        out[base + 32 + l16] = o2[r] * inv;
        out[base + 48 + l16] = o3[r] * inv;
    }
}

// ---------------------------------------------------------------------------
extern "C" void kernel_launch(void* const* d_in, const int* in_sizes, int n_in,
                              void* d_out, int out_size, void* d_ws, size_t ws_size,
                              hipStream_t stream)
{
    (void)in_sizes; (void)n_in; (void)out_size; (void)ws_size;
    const float* x      = (const float*)d_in[0];
    const float* past_k = (const float*)d_in[1];
    const float* past_v = (const float*)d_in[2];
    const float* imp    = (const float*)d_in[3];
    const float* qkv_w  = (const float*)d_in[4];
    const float* qkv_b  = (const float*)d_in[5];
    const float* proj_w = (const float*)d_in[6];
    const float* proj_b = (const float*)d_in[7];
    float* out = (float*)d_out;
    float* ws  = (float*)d_ws;

    // workspace layout (float indices; ~91 MB total)
    const size_t SZ_QKV = (size_t)B_ * H_ * N_ * HD_;        // 2,097,152
    const size_t SZ_KP  = (size_t)B_ * H_ * KEEP_ * HD_;     // 7,143,424
    float* q      = ws;
    float* knew   = q + SZ_QKV;
    float* vnew   = knew + SZ_QKV;
    float* kkeep  = vnew + SZ_QKV;
    float* vkeep  = kkeep + SZ_KP;
    float* attno  = vkeep + SZ_KP;                           // (B,N,C)
    float* knmean = attno + (size_t)B_ * N_ * C_;            // B*H*HD
    float* divavg = knmean + (size_t)B_ * H_ * HD_;          // B*NOLD_
    float* minmax = divavg + (size_t)B_ * NOLD_;             // B*8
    float* comb   = minmax + (size_t)B_ * 8;                 // B*NC_
    int*   flag   = (int*)(comb + (size_t)B_ * NC_);         // B*NC_
    int*   gidx   = flag + (size_t)B_ * NC_;                 // B*KEEP_

    // 1) QKV projection (2048 x 3072 x 1024), scatter to q/knew/vnew
    k_gemm_nt<<<6144, 128, 0, stream>>>(x, qkv_w, qkv_b, nullptr, q, knew, vnew,
                                        B_ * N_, 3 * C_, C_, 1);
    // 2) mean of normalized candidate keys per (b,h)
    k_knmean<<<B_ * H_, 256, 0, stream>>>(past_k, knew, knmean);
    // 3) diversity score for old candidates
    k_divavg<<<(B_ * NOLD_) / 8, 256, 0, stream>>>(past_k, knmean, divavg);
    // 4) per-batch min/max of diversity + importance
    k_minmax<<<B_, 256, 0, stream>>>(divavg, imp, minmax);
    // 5) blended score
    k_comb<<<(B_ * NC_) / 256, 256, 0, stream>>>(divavg, imp, minmax, comb);
    // 6) exact top-k membership by rank counting
    k_rank<<<(B_ * NC_) / 256, 256, 0, stream>>>(comb, flag);
    // 7) compact kept indices in ascending order
    k_compact<<<B_, 32, 0, stream>>>(flag, gidx);
    // 8) gather compact K/V caches
    k_gather<<<(unsigned)(((size_t)B_ * H_ * KEEP_ * 16) / 256), 256, 0, stream>>>(
        past_k, past_v, knew, vnew, gidx, kkeep, vkeep);
    // 9) flash attention over anchor + kept keys
    k_attn<<<(B_ * H_ * (N_ / 16)) / 4, 128, 0, stream>>>(q, past_k, past_v, kkeep, vkeep, attno);
    // 10) output projection (2048 x 1024 x 1024)
    k_gemm_nt<<<2048, 128, 0, stream>>>(attno, proj_w, proj_b, out,
                                        nullptr, nullptr, nullptr, B_ * N_, C_, C_, 0);
}